// Dynamic_Relu_B_70291434766473
// MI455X (gfx1250) — compile-verified
//
#include <hip/hip_runtime.h>
#include <math.h>

typedef __attribute__((ext_vector_type(2))) float v2f;
typedef __attribute__((ext_vector_type(4))) float v4f;
typedef __attribute__((ext_vector_type(8))) float v8f;

#define NB   32      // batch
#define CC   256     // channels
#define HIDN 32      // hidden = C/8
#define HW   3136    // 56*56
#define HW4  784     // HW/4
#define OO   512     // 2*C
#define LDSS 34      // LDS row stride for h (even -> aligned float2, conflict-free)

// ---------------- Kernel 1: mean pool over H*W, one block per (n,c) plane ----
__global__ __launch_bounds__(256) void pool_kernel(const float* __restrict__ x,
                                                   float* __restrict__ pooled) {
  const int bid = blockIdx.x;                       // n*C + c
  const v4f* p4 = (const v4f*)(x + (size_t)bid * HW);
  float s = 0.f;
  for (int i = threadIdx.x; i < HW4; i += 256) {
    v4f v = p4[i];
    s += (v.x + v.y) + (v.z + v.w);
  }
  // wave32 tree reduction
  for (int off = 16; off > 0; off >>= 1)
    s += __shfl_down(s, off, 32);
  __shared__ float wsum[8];
  const int lane = threadIdx.x & 31, wave = threadIdx.x >> 5;
  if (lane == 0) wsum[wave] = s;
  __syncthreads();
  if (threadIdx.x == 0) {
    float t = 0.f;
    #pragma unroll
    for (int w = 0; w < 8; ++w) t += wsum[w];
    pooled[bid] = t * (1.0f / (float)HW);
  }
}

// ---------------- Kernel 2: coefficient chain, single workgroup, f32 WMMA ----
// Phase A: h = relu(pooled(32x256) @ fc1_w^T(256x32) + b1)   -> LDS
// Phase B: g[k] = h(32x32) @ fc2_w[k]^T(32x512) + b2[k]; delta = tanh(g/2)
//          a[k,n,c] = init_a[k] + delta_even ; b[k,n,c] = init_b[k] + 0.5*delta_odd
__global__ __launch_bounds__(256) void coef_kernel(const float* __restrict__ pooled,
    const float* __restrict__ fc1_w, const float* __restrict__ fc1_b,
    const float* __restrict__ fc2_w, const float* __restrict__ fc2_b,
    float* __restrict__ Acoef, float* __restrict__ Bcoef) {
  __shared__ float Hs[NB * LDSS];

  const int tid  = threadIdx.x;
  const int lane = tid & 31;
  const int wave = tid >> 5;
  const int hl   = lane >> 4;        // lane half: selects K pair per 16x16x4 layout
  const int l16  = lane & 15;

  // ---- Phase A: 4 waves, one 16x16 tile each (2x2 tiling of 32x32), K=256 ----
  if (wave < 4) {
    const int tm = wave >> 1, tn = wave & 1;
    const int m  = tm * 16 + l16;          // A-matrix row (pooled batch row)
    const int nc = tn * 16 + l16;          // B-matrix col (hidden unit)
    v8f acc = {};
    #pragma unroll 8
    for (int k = 0; k < CC; k += 4) {
      // A 16x4 f32 frag: lane m holds K = k + 2*hl, +1  (consecutive -> float2)
      v2f a = *(const v2f*)(pooled + m * CC + k + 2 * hl);
      // B 4x16 f32 frag: B[kk][nc] = fc1_w[nc][kk] (row-major in C -> float2)
      v2f b = *(const v2f*)(fc1_w + nc * CC + k + 2 * hl);
      acc = __builtin_amdgcn_wmma_f32_16x16x4_f32(false, a, false, b,
                                                  (short)0, acc, false, false);
    }
    const float bias = fc1_b[nc];
    #pragma unroll
    for (int j = 0; j < 8; ++j) {
      const int n = tm * 16 + j + 8 * hl;  // C/D layout: VGPR j, lane-half
      float h = acc[j] + bias;
      Hs[n * LDSS + nc] = h > 0.f ? h : 0.f;
    }
  }
  __syncthreads();

  // ---- Phase B: 8 waves sweep 2(k) * 2(n-tiles) * 32(o-tiles) = 128 tiles ----
  for (int t = wave; t < 128; t += 8) {
    const int k2  = t >> 6;
    const int rem = t & 63;
    const int tm  = rem & 1;           // n tile
    const int to  = rem >> 1;          // o tile (0..31)
    const int m   = tm * 16 + l16;
    const int o   = to * 16 + l16;
    const float* wrow = fc2_w + ((size_t)k2 * OO + o) * HIDN;
    v8f acc = {};
    #pragma unroll
    for (int h = 0; h < HIDN; h += 4) {
      v2f a = *(const v2f*)(&Hs[m * LDSS + h + 2 * hl]);   // ds_load_b64
      v2f b = *(const v2f*)(wrow + h + 2 * hl);
      acc = __builtin_amdgcn_wmma_f32_16x16x4_f32(false, a, false, b,
                                                  (short)0, acc, false, false);
    }
    const float bias = fc2_b[k2 * OO + o];
    const float init = (k2 == 0) ? 1.f : 0.f;   // init_alpha == init_beta == [1,0]
    const int   cidx = o >> 1;
    #pragma unroll
    for (int j = 0; j < 8; ++j) {
      const int n = tm * 16 + j + 8 * hl;
      float g = acc[j] + bias;
      float e = __expf(-g);
      float d = (1.f - e) / (1.f + e);          // 2*sigmoid(g) - 1 == tanh(g/2)
      if (o & 1) Bcoef[(k2 * NB + n) * CC + cidx] = init + 0.5f * d;
      else       Acoef[(k2 * NB + n) * CC + cidx] = init + d;
    }
  }
}

// ---------------- Kernel 3: streaming epilogue, one block per (n,c) plane ----
__global__ __launch_bounds__(256) void fuse_kernel(const float* __restrict__ x,
    const float* __restrict__ Acoef, const float* __restrict__ Bcoef,
    float* __restrict__ out) {
  const int bid = blockIdx.x;                    // n*C + c
  const float a0 = Acoef[bid],           b0 = Bcoef[bid];
  const float a1 = Acoef[NB * CC + bid], b1 = Bcoef[NB * CC + bid];
  const v4f* xp = (const v4f*)(x   + (size_t)bid * HW);
  v4f*       op = (v4f*)      (out + (size_t)bid * HW);
  for (int i = threadIdx.x; i < HW4; i += 256) {
    v4f v = xp[i];                               // regular load: hits L2 (x < 192MB)
    v4f r;
    r.x = fmaxf(fmaf(v.x, a0, b0), fmaf(v.x, a1, b1));
    r.y = fmaxf(fmaf(v.y, a0, b0), fmaf(v.y, a1, b1));
    r.z = fmaxf(fmaf(v.z, a0, b0), fmaf(v.z, a1, b1));
    r.w = fmaxf(fmaf(v.w, a0, b0), fmaf(v.w, a1, b1));
    __builtin_nontemporal_store(r, op + i);      // NT store: keep x resident in L2
  }
}

extern "C" void kernel_launch(void* const* d_in, const int* in_sizes, int n_in,
                              void* d_out, int out_size, void* d_ws, size_t ws_size,
                              hipStream_t stream) {
  const float* x     = (const float*)d_in[0];
  const float* fc1_w = (const float*)d_in[1];
  const float* fc1_b = (const float*)d_in[2];
  const float* fc2_w = (const float*)d_in[3];
  const float* fc2_b = (const float*)d_in[4];
  float* out = (float*)d_out;

  float* pooled = (float*)d_ws;            // 32*256           =  8192 f
  float* Acoef  = pooled + NB * CC;        // 2*32*256         = 16384 f
  float* Bcoef  = Acoef + 2 * NB * CC;     // 2*32*256         = 16384 f
                                           // total 160 KB of d_ws

  pool_kernel<<<NB * CC, 256, 0, stream>>>(x, pooled);
  coef_kernel<<<1, 256, 0, stream>>>(pooled, fc1_w, fc1_b, fc2_w, fc2_b, Acoef, Bcoef);
  fuse_kernel<<<NB * CC, 256, 0, stream>>>(x, Acoef, Bcoef, out);
}